// GTOutputDoubleWindowPooling_63677185131071
// MI455X (gfx1250) — compile-verified
//
#include <hip/hip_runtime.h>
#include <hip/hip_bf16.h>

// ---------------------------------------------------------------------------
// GTOutputDoubleWindowPooling for MI455X (gfx1250, wave32)
//
// Phase 1 (memory bound, ~640MB @ 23.3TB/s ~= 27us floor):
//   k_part   : chunked streaming max+sum over s=1..S-1, float4 nontemporal
//   k_final  : deterministic combine of chunk partials -> gmax/gavg
// Phase 2 (tiny):
//   k_denom  : denom0[b] = sum(mask[b,:]) - 1
//   k_feat   : gather gap rows + +/-12 window pools -> F (160 x 6400)
//   k_wprep  : pack weights into 2 contiguous columns (branchless B feed)
//   k_logits : F @ [w_normal | w_start] via V_WMMA_F32_16X16X4_F32.
//              K tiles staged global->LDS with GLOBAL_LOAD_ASYNC_TO_LDS_B128
//              (ASYNCcnt, zero VGPR staging), 4 waves/block split each tile's
//              K, wmma fed from LDS, deterministic LDS combine.
// ---------------------------------------------------------------------------

#define B_     32
#define S_     4096
#define DA_    1024
#define DM_    256
#define WIN_   12
#define NCHUNK 16
#define ROWSPC 256            // ceil((S_-1)/NCHUNK)
#define KTOT   6400           // 5*DA_ + 5*DM_
#define KATT   5120           // 5*DA_
#define KT_    256            // K tile staged in LDS (25 tiles)
#define FMAXF  3.402823466e+38f

typedef float f4  __attribute__((ext_vector_type(4)));
typedef float v2f __attribute__((ext_vector_type(2)));
typedef float v8f __attribute__((ext_vector_type(8)));

__device__ __forceinline__ f4 fmax4(f4 a, f4 b) {
    f4 r;
    r.x = fmaxf(a.x, b.x); r.y = fmaxf(a.y, b.y);
    r.z = fmaxf(a.z, b.z); r.w = fmaxf(a.w, b.w);
    return r;
}
__device__ __forceinline__ int imin(int a, int b) { return a < b ? a : b; }
__device__ __forceinline__ int imax(int a, int b) { return a > b ? a : b; }

// gfx1250 async copy: 16B global -> LDS, no VGPR staging, tracked by ASYNCcnt.
// VDST operand = LDS byte address (low 32 bits of generic pointer to shared).
__device__ __forceinline__ void async_copy_b128(const void* gaddr, void* lds_ptr) {
    const unsigned lds_addr = (unsigned)(size_t)lds_ptr;
    asm volatile("global_load_async_to_lds_b128 %0, %1, off"
                 :: "v"(lds_addr), "v"(gaddr)
                 : "memory");
}
__device__ __forceinline__ void wait_asynccnt0() {
    asm volatile("s_wait_asynccnt 0x0" ::: "memory");
}

// --------------------------- denom0[b] = sum(mask[b,:]) - 1 -----------------
__global__ __launch_bounds__(256) void k_denom(const float* __restrict__ mask,
                                               float* __restrict__ denom0) {
    __shared__ float sm[256];
    const int b = blockIdx.x;
    float s = 0.f;
    for (int i = threadIdx.x; i < S_; i += 256) s += mask[(size_t)b * S_ + i];
    sm[threadIdx.x] = s;
    __syncthreads();
    for (int st = 128; st > 0; st >>= 1) {
        if (threadIdx.x < st) sm[threadIdx.x] += sm[threadIdx.x + st];
        __syncthreads();
    }
    if (threadIdx.x == 0) denom0[b] = sm[0] - 1.0f;
}

// ------------------ chunked streaming reduction over s=1..S-1 ---------------
// grid (B_, NCHUNK), block = D/4 threads; each thread owns one float4 column.
__global__ void k_part(const float* __restrict__ x, int D,
                       float* __restrict__ pmax, float* __restrict__ psum) {
    const int b = blockIdx.x, c = blockIdx.y;
    const int t = threadIdx.x;                 // t < D/4
    const int stride = D >> 2;
    const int s0 = 1 + c * ROWSPC;
    const int s1 = imin(S_, s0 + ROWSPC);
    const f4* __restrict__ xb = (const f4*)(x + (size_t)b * S_ * D);
    f4 vmax = { -FMAXF, -FMAXF, -FMAXF, -FMAXF };
    f4 vsum = { 0.f, 0.f, 0.f, 0.f };
#pragma unroll 8
    for (int s = s0; s < s1; ++s) {
        f4 v = __builtin_nontemporal_load(&xb[(size_t)s * stride + t]);
        vmax = fmax4(vmax, v);
        vsum += v;
    }
    const size_t o = ((size_t)b * NCHUNK + c) * (size_t)stride + t;
    ((f4*)pmax)[o] = vmax;
    ((f4*)psum)[o] = vsum;
}

// ------------------ combine chunk partials, divide by denom0 ----------------
__global__ __launch_bounds__(256) void k_final(const float* __restrict__ pmax,
                                               const float* __restrict__ psum,
                                               const float* __restrict__ denom0,
                                               float* __restrict__ gmax,
                                               float* __restrict__ gavg, int D) {
    const int b = blockIdx.x;
    const int d = blockIdx.y * 256 + threadIdx.x;
    if (d >= D) return;
    float m = -FMAXF, s = 0.f;
#pragma unroll
    for (int c = 0; c < NCHUNK; ++c) {
        const size_t o = ((size_t)b * NCHUNK + c) * (size_t)D + d;
        m = fmaxf(m, pmax[o]);
        s += psum[o];
    }
    gmax[(size_t)b * D + d] = m;
    gavg[(size_t)b * D + d] = s / denom0[b];
}

// ------------------ build feature rows F[r=b*5+i][6400] ---------------------
__device__ __forceinline__ void feat_stream(const float* __restrict__ x, int D,
                                            int b, int i, int gi,
                                            const int* __restrict__ gaps,
                                            const float* __restrict__ mask,
                                            const float* __restrict__ gmax,
                                            const float* __restrict__ gavg,
                                            float* __restrict__ Fr) {
    const float* __restrict__ xb = x + (size_t)b * S_ * D;
    const float* __restrict__ mb = mask + (size_t)b * S_;
    for (int d = threadIdx.x; d < D; d += blockDim.x) {
        const float gapv = xb[(size_t)gi * D + d];
        float p1mx, p1av, p2mx, p2av;
        if (i == 0) {
            p1mx = gmax[(size_t)b * D + d];
            p1av = gavg[(size_t)b * D + d];
            float m2 = -FMAXF, s2 = 0.f;
            for (int j = 1; j < 5; ++j) {
                const float v = xb[(size_t)gaps[b * 5 + j] * D + d];
                m2 = fmaxf(m2, v);
                s2 += v;
            }
            p2mx = m2;
            p2av = s2 * 0.25f;
        } else {
            float mx1 = -FMAXF, s1 = 0.f, w1 = 0.f;
            for (int o = -WIN_; o <= -1; ++o) {          // offs1 ascending
                const int p = gi + o;
                const int pc = imin(imax(p, 0), S_ - 1);
                const float wm = (p >= 0 && p < S_) ? mb[pc] : 0.f;
                const float v = xb[(size_t)pc * D + d] * wm;
                mx1 = fmaxf(mx1, v); s1 += v; w1 += wm;
            }
            float mx2 = -FMAXF, s2 = 0.f, w2 = 0.f;
            for (int o = 1; o <= WIN_; ++o) {            // offs2 ascending
                const int p = gi + o;
                const int pc = imin(imax(p, 0), S_ - 1);
                const float wm = (p >= 0 && p < S_) ? mb[pc] : 0.f;
                const float v = xb[(size_t)pc * D + d] * wm;
                mx2 = fmaxf(mx2, v); s2 += v; w2 += wm;
            }
            p1mx = fmaxf(mx1, 0.f); p1av = s1 / w1;
            p2mx = fmaxf(mx2, 0.f); p2av = s2 / w2;
        }
        Fr[d]         = gapv;
        Fr[D + d]     = p1mx;
        Fr[2 * D + d] = p1av;
        Fr[3 * D + d] = p2mx;
        Fr[4 * D + d] = p2av;
    }
}

__global__ __launch_bounds__(256) void k_feat(const float* __restrict__ att,
                                              const float* __restrict__ mod,
                                              const int* __restrict__ gaps,
                                              const float* __restrict__ mask,
                                              const float* __restrict__ gmax_att,
                                              const float* __restrict__ gavg_att,
                                              const float* __restrict__ gmax_mod,
                                              const float* __restrict__ gavg_mod,
                                              float* __restrict__ F) {
    const int r = blockIdx.x;         // 0..159
    const int b = r / 5, i = r % 5;
    const int gi = gaps[b * 5 + i];
    float* Fr = F + (size_t)r * KTOT;
    feat_stream(att, DA_, b, i, gi, gaps, mask, gmax_att, gavg_att, Fr);
    feat_stream(mod, DM_, b, i, gi, gaps, mask, gmax_mod, gavg_mod, Fr + KATT);
}

// ------------- pack weights: wc[0][k]=normal, wc[1][k]=start ---------------
__global__ __launch_bounds__(256) void k_wprep(const float* __restrict__ w_att,
                                               const float* __restrict__ w_mod,
                                               const float* __restrict__ w_att_s,
                                               const float* __restrict__ w_mod_s,
                                               float* __restrict__ wc) {
    const int idx = blockIdx.x * 256 + threadIdx.x;
    if (idx >= 2 * KTOT) return;
    const int col = idx / KTOT, k = idx % KTOT;
    const float* a = col ? w_att_s : w_att;
    const float* m = col ? w_mod_s : w_mod;
    wc[idx] = (k < KATT) ? a[k] : m[k - KATT];
}

// ------------------ logits via V_WMMA_F32_16X16X4_F32 -----------------------
// P(160x16) = F(160x6400) @ W(6400x16); W col0 = normal weights, col1 = start
// weights, cols 2..15 = 0. logits[row] = P[row, (row%5==0)?1:0] + biases.
// Per 256-wide K tile: async global->LDS stage (9 in-flight copies/thread,
// one ASYNCcnt wait), then each of 4 waves runs 16 wmmas on its 64-wide
// quarter fed from LDS. Partial D tiles combined deterministically via LDS.
// Layouts per CDNA5 ISA 7.12.2:
//   A 16x4 f32 : lane%16 = M; v0 holds K=0/2, v1 K=1/3 (by lane half)
//   B 4x16 f32 : lane%16 = N; lane-half selects K pair {0,1}/{2,3}
//   C/D 16x16  : lane%16 = N; VGPR v -> M = (lane/16)*8 + v
__global__ __launch_bounds__(128) void k_logits(const float* __restrict__ F,
                                                const float* __restrict__ wc,
                                                const float* __restrict__ b_att,
                                                const float* __restrict__ b_mod,
                                                const float* __restrict__ b_att_s,
                                                const float* __restrict__ b_mod_s,
                                                float* __restrict__ out) {
    __shared__ float Ftile[16][KT_];        // 16 KB
    __shared__ float Wtile[2][KT_];         // 2 KB
    __shared__ float tiles[4][16][16];      // 4 KB, wave partials
    const int tid  = threadIdx.x;           // 0..127
    const int wave = tid >> 5;              // 0..3: K quarter of each tile
    const int lane = tid & 31;
    const int half = lane >> 4;
    const int l16  = lane & 15;
    const int rowBase = blockIdx.x * 16;
    const float wscale = (l16 < 2) ? 1.f : 0.f;   // B cols 2..15 are zero

    v8f acc = { 0.f, 0.f, 0.f, 0.f, 0.f, 0.f, 0.f, 0.f };
    for (int kt = 0; kt < KTOT; kt += KT_) {      // 25 tiles
        // ---- async stage: 8 + 1 concurrent 16B copies per thread ----
#pragma unroll
        for (int j = 0; j < (16 * KT_ / 4) / 128; ++j) {   // 8
            const int idx = tid + j * 128;                 // f4 index
            const int row = idx >> 6;                      // KT_/4 = 64 per row
            const int c4  = idx & 63;
            async_copy_b128(F + (size_t)(rowBase + row) * KTOT + kt + c4 * 4,
                            &Ftile[row][c4 * 4]);
        }
        {
            const int col = tid >> 6, c4 = tid & 63;       // 2*64 f4 groups
            async_copy_b128(wc + (size_t)col * KTOT + kt + c4 * 4,
                            &Wtile[col][c4 * 4]);
        }
        wait_asynccnt0();
        __syncthreads();
        // ---- wave's 64-wide K quarter: 16 LDS-fed wmma steps ----
        const int kb = wave * 64;
        v2f a[16], bb[16];
#pragma unroll
        for (int u = 0; u < 16; ++u) {
            const int ka = kb + 4 * u + half * 2;          // 8B-aligned
            a[u]  = *(const v2f*)(&Ftile[l16][ka]);
            bb[u] = wscale * (*(const v2f*)(&Wtile[l16 & 1][ka]));
        }
#pragma unroll
        for (int u = 0; u < 16; ++u) {
            acc = __builtin_amdgcn_wmma_f32_16x16x4_f32(false, a[u], false, bb[u],
                                                        (short)0, acc, false, false);
        }
        __syncthreads();
    }
#pragma unroll
    for (int v = 0; v < 8; ++v) tiles[wave][half * 8 + v][l16] = acc[v];
    __syncthreads();
    if (tid < 16) {
        const int row = rowBase + tid;
        const int i = row % 5;
        const int col = (i == 0) ? 1 : 0;
        const float bias = (i == 0) ? (b_att_s[0] + b_mod_s[0])
                                    : (b_att[0] + b_mod[0]);
        const float val = tiles[0][tid][col] + tiles[1][tid][col] +
                          tiles[2][tid][col] + tiles[3][tid][col];
        out[row] = val + bias;
    }
}

// ---------------------------------------------------------------------------
extern "C" void kernel_launch(void* const* d_in, const int* in_sizes, int n_in,
                              void* d_out, int out_size, void* d_ws, size_t ws_size,
                              hipStream_t stream) {
    const float* att     = (const float*)d_in[0];
    const float* mod_    = (const float*)d_in[1];
    const int*   gaps    = (const int*)d_in[2];
    const float* mask    = (const float*)d_in[3];
    const float* w_att   = (const float*)d_in[4];
    const float* b_att   = (const float*)d_in[5];
    const float* w_mod   = (const float*)d_in[6];
    const float* b_mod   = (const float*)d_in[7];
    const float* w_att_s = (const float*)d_in[8];
    const float* b_att_s = (const float*)d_in[9];
    const float* w_mod_s = (const float*)d_in[10];
    const float* b_mod_s = (const float*)d_in[11];
    float* out = (float*)d_out;

    // workspace layout (floats), total ~= 9.7 MB
    float* ws = (float*)d_ws;
    float* att_pmax = ws;                 ws += (size_t)B_ * NCHUNK * DA_;
    float* att_psum = ws;                 ws += (size_t)B_ * NCHUNK * DA_;
    float* mod_pmax = ws;                 ws += (size_t)B_ * NCHUNK * DM_;
    float* mod_psum = ws;                 ws += (size_t)B_ * NCHUNK * DM_;
    float* gmax_att = ws;                 ws += (size_t)B_ * DA_;
    float* gavg_att = ws;                 ws += (size_t)B_ * DA_;
    float* gmax_mod = ws;                 ws += (size_t)B_ * DM_;
    float* gavg_mod = ws;                 ws += (size_t)B_ * DM_;
    float* denom0   = ws;                 ws += B_;
    float* Fbuf     = ws;                 ws += (size_t)B_ * 5 * KTOT;
    float* wcomb    = ws;                 ws += (size_t)2 * KTOT;

    k_wprep<<<(2 * KTOT + 255) / 256, 256, 0, stream>>>(w_att, w_mod, w_att_s,
                                                        w_mod_s, wcomb);
    k_denom<<<B_, 256, 0, stream>>>(mask, denom0);
    k_part<<<dim3(B_, NCHUNK), DA_ / 4, 0, stream>>>(att, DA_, att_pmax, att_psum);
    k_part<<<dim3(B_, NCHUNK), DM_ / 4, 0, stream>>>(mod_, DM_, mod_pmax, mod_psum);
    k_final<<<dim3(B_, DA_ / 256), 256, 0, stream>>>(att_pmax, att_psum, denom0,
                                                     gmax_att, gavg_att, DA_);
    k_final<<<dim3(B_, 1), 256, 0, stream>>>(mod_pmax, mod_psum, denom0,
                                             gmax_mod, gavg_mod, DM_);
    k_feat<<<B_ * 5, 256, 0, stream>>>(att, mod_, gaps, mask, gmax_att, gavg_att,
                                       gmax_mod, gavg_mod, Fbuf);
    k_logits<<<(B_ * 5) / 16, 128, 0, stream>>>(Fbuf, wcomb, b_att, b_mod,
                                                b_att_s, b_mod_s, out);
}